// Quanv3x3LayerClass_25812753449813
// MI455X (gfx1250) — compile-verified
//
#include <hip/hip_runtime.h>
#include <math.h>

typedef __attribute__((ext_vector_type(2))) float v2f;
typedef __attribute__((ext_vector_type(8))) float v8f;

#define NQ     9
#define NSTATE 512          // 2^9 amplitudes
#define ROWS   256          // rows with qubit0 == 0
#define KDIM   512          // GEMM inner dim (basis columns)
#define P_TOT  3600         // 4 * 30 * 30 patches
#define COUT   4
#define NPAR   16
#define CB     225          // 3600 / 16 column blocks

#define U2_ELEMS (COUT * 2 * ROWS * KDIM)   // 1,048,576 floats (4 MB)
#define TRIG_ELEMS (P_TOT * NQ)             // 32,400 floats each (cos, sin)

// circuit: (ctrl, tgt, param-base k) ; gate g uses params k=2g (RZ) and 2g+1 (RX)
__device__ __constant__ int GCTRL[8] = {1, 3, 2, 8, 5, 7, 6, 4};
__device__ __constant__ int GTGT [8] = {0, 2, 0, 7, 4, 6, 4, 0};

// ---------------------------------------------------------------------------
// Kernel 1: per-patch, per-qubit cos/sin of half-angles.
//   p in [0,3600): p = b*900 + i*30 + j ; qubit q = 3*di + dj uses pixel
//   (i+di, j+dj). angle = 2*pi*mean_c(x) - pi ; store cos(angle/2), sin(angle/2).
// ---------------------------------------------------------------------------
__global__ void quanv_trig_kernel(const float* __restrict__ x,
                                  float* __restrict__ ctab,
                                  float* __restrict__ stab) {
    int idx = blockIdx.x * blockDim.x + threadIdx.x;
    if (idx >= P_TOT * NQ) return;
    int p = idx / NQ;
    int q = idx % NQ;
    int b   = p / 900;
    int rem = p % 900;
    int i0  = rem / 30;
    int j0  = rem % 30;
    int h = i0 + q / 3;
    int w = j0 + q % 3;
    const float* px = x + ((b * 32 + h) * 32 + w) * 3;
    float avg = (px[0] + px[1] + px[2]) * (1.0f / 3.0f);
    float ha = 3.14159265358979323846f * avg - 1.57079632679489661923f; // angle/2
    ctab[idx] = cosf(ha);
    stab[idx] = sinf(ha);
}

// ---------------------------------------------------------------------------
// Kernel 2: build U'' (rows of the circuit unitary with qubit0 == 0), with
// the (-i)^popcount(col) column phase folded in. One block per (cout, col).
// State (512 complex) lives in LDS; 256 threads apply the 16 gates.
// Bit q of an amplitude index == qubit q.
// ---------------------------------------------------------------------------
__global__ void quanv_unitary_kernel(const float* __restrict__ qp,
                                     float* __restrict__ U2) {
    __shared__ float sre[NSTATE];
    __shared__ float sim[NSTATE];

    const int col  = blockIdx.x;   // basis column 0..511
    const int cout = blockIdx.y;   // 0..3
    const int t    = threadIdx.x;  // 0..255

    // init |col>
    sre[t] = 0.0f;        sim[t] = 0.0f;
    sre[t + 256] = 0.0f;  sim[t + 256] = 0.0f;
    __syncthreads();
    if (t == 0) sre[col] = 1.0f;

    #pragma unroll
    for (int g = 0; g < 8; ++g) {
        const int ctrl = GCTRL[g];
        const int tgt  = GTGT[g];
        const float thz = qp[cout * NPAR + 2 * g];
        const float thx = qp[cout * NPAR + 2 * g + 1];
        const float cz = cosf(0.5f * thz), sz = sinf(0.5f * thz);
        const float cx = cosf(0.5f * thx), sx = sinf(0.5f * thx);

        // ---- controlled RZ: diagonal on the 256 amplitudes with ctrl=1 ----
        __syncthreads();
        {
            int low = t & ((1 << ctrl) - 1);
            int n = ((t >> ctrl) << (ctrl + 1)) | (1 << ctrl) | low;
            float s = ((n >> tgt) & 1) ? sz : -sz;   // e^{+-i th/2}
            float re = sre[n], im = sim[n];
            sre[n] = re * cz - im * s;
            sim[n] = re * s + im * cz;
        }
        __syncthreads();

        // ---- controlled RX: 128 pairs (ctrl=1), mix tgt=0 <-> tgt=1 ----
        if (t < 128) {
            int lo = ctrl < tgt ? ctrl : tgt;
            int hi = ctrl < tgt ? tgt : ctrl;
            int m1 = ((t >> lo) << (lo + 1)) | (t & ((1 << lo) - 1));
            int i0 = ((m1 >> hi) << (hi + 1)) | (m1 & ((1 << hi) - 1));
            int i = i0 | (1 << ctrl);   // tgt bit = 0
            int j = i | (1 << tgt);
            float ri = sre[i], ii = sim[i];
            float rj = sre[j], ij = sim[j];
            sre[i] = cx * ri + sx * ij;
            sim[i] = cx * ii - sx * rj;
            sre[j] = cx * rj + sx * ii;
            sim[j] = cx * ij - sx * ri;
        }
        __syncthreads();
    }

    // fold column phase (-i)^popcount(col); keep rows with bit0 == 0 (n = 2m)
    const int k = __popc(col) & 3;
    const int m = t;
    const int n = m << 1;
    float re = sre[n], im = sim[n];
    float pr, pi;
    switch (k) {
        case 0: pr = re;  pi = im;  break;
        case 1: pr = im;  pi = -re; break;
        case 2: pr = -re; pi = -im; break;
        default: pr = -im; pi = re; break;
    }
    U2[((cout * 2 + 0) * ROWS + m) * KDIM + col] = pr;
    U2[((cout * 2 + 1) * ROWS + m) * KDIM + col] = pi;
}

// ---------------------------------------------------------------------------
// Kernel 3: fused WMMA GEMM + |.|^2 row-reduction.
//   Y_c = U''_c (256x512) * v (512x16 strip) for c in {re, im};
//   p0[p] = sum over 256 rows of Y_re^2 + Y_im^2.
// One 128-thread block (4 waves) per (16-patch strip, cout).
// v strip is built in LDS from the trig table:
//   v[k][n] = prod_q (bit_q(k) ? sin : cos)(patch n, qubit q)   (real!)
// ---------------------------------------------------------------------------
__global__ void quanv_wmma_kernel(const float* __restrict__ U2,
                                  const float* __restrict__ ctab,
                                  const float* __restrict__ stab,
                                  float* __restrict__ out) {
    __shared__ float Bs[KDIM * 16];      // 32 KB: v strip, [k][n]
    __shared__ float cT[16 * NQ];
    __shared__ float sT[16 * NQ];
    __shared__ float partial[4 * 16];

    const int cb   = blockIdx.x;   // 0..224
    const int cout = blockIdx.y;   // 0..3
    const int tid  = threadIdx.x;  // 0..127
    const int lane = tid & 31;
    const int wave = tid >> 5;
    const int half = lane >> 4;    // 0: lanes 0-15, 1: lanes 16-31
    const int lq   = lane & 15;

    // stage trig values for this 16-patch strip
    if (tid < 16 * NQ) {
        int pl = tid / NQ, q = tid % NQ;
        int p = cb * 16 + pl;
        cT[tid] = ctab[p * NQ + q];
        sT[tid] = stab[p * NQ + q];
    }
    __syncthreads();

    // build the real amplitude strip v: 512 x 16
    for (int idx = tid; idx < KDIM * 16; idx += 128) {
        int kk = idx >> 4;
        int n  = idx & 15;
        float prod = 1.0f;
        #pragma unroll
        for (int q = 0; q < NQ; ++q)
            prod *= ((kk >> q) & 1) ? sT[n * NQ + q] : cT[n * NQ + q];
        Bs[kk * 16 + n] = prod;
    }
    __syncthreads();

    // GEMM: each wave owns row tiles {wave, wave+4, wave+8, wave+12}
    float sumsq = 0.0f;
    for (int rt = wave; rt < 16; rt += 4) {
        #pragma unroll
        for (int comp = 0; comp < 2; ++comp) {
            const float* Ab = U2 + ((size_t)((cout * 2 + comp) * ROWS + rt * 16 + lq)) * KDIM
                              + 2 * half;
            v8f c = {0.f, 0.f, 0.f, 0.f, 0.f, 0.f, 0.f, 0.f};
            #pragma unroll 4
            for (int k0 = 0; k0 < KDIM; k0 += 4) {
                // A tile 16x4 (M x K): lane<16 -> K = k0,k0+1 ; lane>=16 -> K = k0+2,k0+3
                v2f a = *(const v2f*)(Ab + k0);
                // B tile 4x16 (K x N): mirrored halves
                v2f b;
                b.x = Bs[(k0 + 2 * half) * 16 + lq];
                b.y = Bs[(k0 + 2 * half + 1) * 16 + lq];
                c = __builtin_amdgcn_wmma_f32_16x16x4_f32(
                        false, a, false, b, (short)0, c, false, false);
            }
            #pragma unroll
            for (int v = 0; v < 8; ++v) sumsq += c[v] * c[v];
        }
    }

    // columns live at lane lq and lane lq+16: fold halves, then fold waves
    sumsq += __shfl_xor(sumsq, 16, 32);
    if (half == 0) partial[wave * 16 + lq] = sumsq;
    __syncthreads();

    if (tid < 16) {
        float p0 = partial[tid] + partial[16 + tid] + partial[32 + tid] + partial[48 + tid];
        int p   = cb * 16 + tid;
        int b   = p / 900;
        int rem = p % 900;
        int ii  = rem / 30;
        int jj  = rem % 30;
        out[((b * 32 + (ii + 1)) * 32 + (jj + 1)) * COUT + cout] = p0;
    }
}

// ---------------------------------------------------------------------------
extern "C" void kernel_launch(void* const* d_in, const int* in_sizes, int n_in,
                              void* d_out, int out_size, void* d_ws, size_t ws_size,
                              hipStream_t stream) {
    const float* x  = (const float*)d_in[0];   // (4,32,32,3) f32
    const float* qp = (const float*)d_in[1];   // (4,16) f32
    float* out = (float*)d_out;                // (4,32,32,4) f32

    float* ws   = (float*)d_ws;
    float* U2   = ws;                          // 4*2*256*512 floats
    float* ctab = ws + U2_ELEMS;               // 32400 floats
    float* stab = ctab + TRIG_ELEMS;           // 32400 floats

    // output is zero except the 30x30 interior — zero it every call
    hipMemsetAsync(d_out, 0, (size_t)out_size * sizeof(float), stream);

    quanv_trig_kernel<<<(P_TOT * NQ + 255) / 256, 256, 0, stream>>>(x, ctab, stab);

    quanv_unitary_kernel<<<dim3(NSTATE, COUT), 256, 0, stream>>>(qp, U2);

    quanv_wmma_kernel<<<dim3(CB, COUT), 128, 0, stream>>>(U2, ctab, stab, out);
}